// De_conv_batched_multimasks_33672543600808
// MI455X (gfx1250) — compile-verified
//
#include <hip/hip_runtime.h>
#include <stdint.h>

// ---------------------------------------------------------------------------
// Smooth rounding helpers.
// __builtin_amdgcn_sinf(x) == v_sin_f32 == sin(2*pi*x)  (input in revolutions)
// so diff_round(x) = x - sin(2*pi*x)/(2*pi) = fma(-v_sin(x), 1/(2*pi), x)
// ---------------------------------------------------------------------------
#define INV_TWO_PI 0.15915494309189533577f

__device__ __forceinline__ float dround(float x) {
    return __builtin_fmaf(-__builtin_amdgcn_sinf(x), INV_TWO_PI, x);
}
__device__ __forceinline__ float hdr(float x) {          // harder_diff_round
    return dround(dround(dround(x)));
}
// differentiable_eq with the second operand already hardened (bh = hdr(b))
__device__ __forceinline__ float deq_pre(float a, float bh) {
    float ah = hdr(a);
    return hdr(__builtin_fmaf(ah, bh, (1.0f - ah) * (1.0f - bh)));
}
// mask value from 4 equality channels: and(and(c0,c1), and(c2,c3))
__device__ __forceinline__ float mask_of(float c0, float c1, float c2, float c3) {
    float a = dround(c0) * dround(c1);
    float b = dround(c2) * dround(c3);
    return dround(a) * dround(b);
}

// ---------------------------------------------------------------------------
// Async copy: 16 bytes/lane, wave copies 512B per issue; INST_OFFSET is added
// to BOTH the LDS destination and the global source address (ISA 08_async §4.4)
// ---------------------------------------------------------------------------
#define ASYNC_B128(ldsoff, gaddr, OFF)                                        \
    asm volatile("global_load_async_to_lds_b128 %0, %1, off offset:" #OFF     \
                 :: "v"(ldsoff), "v"(gaddr) : "memory")

constexpr int WAVES   = 8;          // waves per block
constexpr int PIX     = 256;        // 16x16 pixels per area
constexpr int TILE_F  = PIX * 4;    // 1024 floats per mask tile (4 channels)

__global__ __launch_bounds__(WAVES * 32)
void De_conv_batched_multimasks_33672543600808_kernel(
        const float* __restrict__ img,    // [A,256,1]
        const float* __restrict__ mA,     // [A,256,4]
        const float* __restrict__ mB,     // [A,256,4]
        const float* __restrict__ mid,    // [A,4]
        float* __restrict__ outA,         // [A,256]
        float* __restrict__ outB,         // [A,256]
        int nAreas)
{
    __shared__ float lds[WAVES * 2 * TILE_F];   // 64 KiB / block

    const int lane = threadIdx.x & 31;
    const int wave = threadIdx.x >> 5;
    const int area = blockIdx.x * WAVES + wave;
    if (area >= nAreas) return;                 // wave-uniform exit, no barriers used

    float* tileA = &lds[(wave * 2 + 0) * TILE_F];
    float* tileB = &lds[(wave * 2 + 1) * TILE_F];

    // Low 32 bits of a flat LDS pointer are the raw LDS byte offset
    // (CDNA5 ISA: flat->LDS aperture keeps addr[31:0] as LDS address).
    const unsigned ldsA = (unsigned)(uintptr_t)tileA + (unsigned)(lane * 16);
    const unsigned ldsB = (unsigned)(uintptr_t)tileB + (unsigned)(lane * 16);

    const uint64_t gA = (uint64_t)(uintptr_t)(mA + (size_t)area * TILE_F) + (uint64_t)(lane * 16);
    const uint64_t gB = (uint64_t)(uintptr_t)(mB + (size_t)area * TILE_F) + (uint64_t)(lane * 16);

    // Stage both 4 KiB mask tiles into LDS via the CDNA5 async path (ASYNCcnt).
    ASYNC_B128(ldsA, gA, 0);    ASYNC_B128(ldsA, gA, 512);
    ASYNC_B128(ldsA, gA, 1024); ASYNC_B128(ldsA, gA, 1536);
    ASYNC_B128(ldsA, gA, 2048); ASYNC_B128(ldsA, gA, 2560);
    ASYNC_B128(ldsA, gA, 3072); ASYNC_B128(ldsA, gA, 3584);
    ASYNC_B128(ldsB, gB, 0);    ASYNC_B128(ldsB, gB, 512);
    ASYNC_B128(ldsB, gB, 1024); ASYNC_B128(ldsB, gB, 1536);
    ASYNC_B128(ldsB, gB, 2048); ASYNC_B128(ldsB, gB, 2560);
    ASYNC_B128(ldsB, gB, 3072); ASYNC_B128(ldsB, gB, 3584);

    // Warm the image line while the async DMA runs (global_prefetch_b8).
    __builtin_prefetch(img + (size_t)area * PIX + lane, 0, 0);

    // Harden the per-area mask id once (saves 12 sins per pixel vs naive).
    const float bh0 = hdr(mid[(size_t)area * 4 + 0]);
    const float bh1 = hdr(mid[(size_t)area * 4 + 1]);
    const float bh2 = hdr(mid[(size_t)area * 4 + 2]);
    const float bh3 = hdr(mid[(size_t)area * 4 + 3]);

    // Wait for both tiles to land in LDS.
    asm volatile("s_wait_asynccnt 0" ::: "memory");

    float mAreg[8], mBreg[8];
    float numA = 0.f, denA = 0.f, numB = 0.f, denB = 0.f;

    #pragma unroll
    for (int i = 0; i < 8; ++i) {
        const int p = lane + 32 * i;
        const float iv = img[(size_t)area * PIX + p];

        const float4 a4 = *reinterpret_cast<const float4*>(tileA + (size_t)p * 4);
        const float4 b4 = *reinterpret_cast<const float4*>(tileB + (size_t)p * 4);

        const float ma = mask_of(deq_pre(a4.x, bh0), deq_pre(a4.y, bh1),
                                 deq_pre(a4.z, bh2), deq_pre(a4.w, bh3));
        const float mb = mask_of(deq_pre(b4.x, bh0), deq_pre(b4.y, bh1),
                                 deq_pre(b4.z, bh2), deq_pre(b4.w, bh3));

        mAreg[i] = ma;  numA = __builtin_fmaf(ma, iv, numA);  denA += ma;
        mBreg[i] = mb;  numB = __builtin_fmaf(mb, iv, numB);  denB += mb;
    }

    // wave32 butterfly reduction
    #pragma unroll
    for (int off = 16; off > 0; off >>= 1) {
        numA += __shfl_xor(numA, off, 32);
        denA += __shfl_xor(denA, off, 32);
        numB += __shfl_xor(numB, off, 32);
        denB += __shfl_xor(denB, off, 32);
    }
    const float meanA = numA / denA;
    const float meanB = numB / denB;

    #pragma unroll
    for (int i = 0; i < 8; ++i) {
        const size_t o = (size_t)area * PIX + lane + 32 * i;
        outA[o] = mAreg[i] * meanA;
        outB[o] = mBreg[i] * meanB;
    }
}

extern "C" void kernel_launch(void* const* d_in, const int* in_sizes, int n_in,
                              void* d_out, int out_size, void* d_ws, size_t ws_size,
                              hipStream_t stream) {
    (void)n_in; (void)out_size; (void)d_ws; (void)ws_size;
    const float* img = (const float*)d_in[0];   // resized_image   [B,N,16,16,1]
    const float* mA  = (const float*)d_in[1];   // mask_combined   [B,N,16,16,4]
    const float* mB  = (const float*)d_in[2];   // mask_combined_alt
    const float* mid = (const float*)d_in[3];   // initial_mask_id [B,N,4]
    // d_in[4] (mask_new_bi_channel) and d_in[5] (mask_index) unused by reference.

    const int nAreas = in_sizes[3] / 4;         // B*N = 16384
    float* outA = (float*)d_out;
    float* outB = outA + (size_t)nAreas * PIX;

    const int blocks = (nAreas + WAVES - 1) / WAVES;
    De_conv_batched_multimasks_33672543600808_kernel<<<blocks, WAVES * 32, 0, stream>>>(
        img, mA, mB, mid, outA, outB, nAreas);
}